// ReluVerifier_8486855377235
// MI455X (gfx1250) — compile-verified
//
#include <hip/hip_runtime.h>

// ---------------------------------------------------------------------------
// ReluVerifier bound propagation, collapsed to closed form.
//
// The reference's diag()/eye()/GEMM structure is degenerate (Ip = I, In = 0,
// all mults diagonal), so per element:
//   slope   = max(ub/(ub-lb), 0)
//   d_ub    = lb>0 ? 1 : slope
//   d_lb    = ub<0 ? 0 : (lb>0 ? 1 : sigmoid(alpha))
//   ub_bias = lb>0 ? 0 : -slope*lb
//   out_ub  = ub_bias + max(d_ub,0)*ub + min(d_ub,0)*lb
//   out_lb  =           max(d_lb,0)*lb + min(d_lb,0)*ub
//
// 80 KB total HBM traffic for N=4096 -> pure latency-bound elementwise kernel.
// CDNA5-specific data path exercised: async global->LDS (ASYNCcnt) staging.
// ---------------------------------------------------------------------------

#define BLK  64   // threads per block (2 wave32)
#define VEC  4    // floats per thread (b128 granularity)
#define TILE (BLK * VEC)

__device__ __forceinline__ void relu_relax(float l, float u, float a,
                                           float& oub, float& olb) {
    float slope = u / (u - l);          // lb < ub guaranteed by verifier
    slope = fmaxf(slope, 0.0f);         // clamp(min=0)
    const bool  pos  = (l > 0.0f);
    const float d_ub = pos ? 1.0f : slope;
    const float sig  = 1.0f / (1.0f + expf(-a));
    const float d_lb = (u < 0.0f) ? 0.0f : (pos ? 1.0f : sig);
    const float ubb  = pos ? 0.0f : (-slope * l);
    oub = ubb + fmaxf(d_ub, 0.0f) * u + fminf(d_ub, 0.0f) * l;
    olb =       fmaxf(d_lb, 0.0f) * l + fminf(d_lb, 0.0f) * u;
}

__global__ __launch_bounds__(BLK)
void relu_verifier_kernel(const float* __restrict__ lb,
                          const float* __restrict__ ub,
                          const float* __restrict__ alpha,
                          float* __restrict__ out, int n) {
    __shared__ __align__(16) float s_lb[TILE];
    __shared__ __align__(16) float s_ub[TILE];
    __shared__ __align__(16) float s_al[TILE];

    const int t    = threadIdx.x;
    const int base = (blockIdx.x * BLK + t) * VEC;   // first element of this lane
    if (base + VEC > n) return;                      // N=4096 divides evenly; tail guard

    // --- CDNA5 async global->LDS staging (ASYNCcnt path) -------------------
    // Each lane copies its own 16B slice of lb/ub/alpha into its own LDS slot,
    // then drains ASYNCcnt. No cross-lane LDS sharing -> no barrier required.
    {
        const unsigned llb = (unsigned)(size_t)(&s_lb[t * VEC]); // low 32 bits of
        const unsigned lub = (unsigned)(size_t)(&s_ub[t * VEC]); // flat shared ptr
        const unsigned lal = (unsigned)(size_t)(&s_al[t * VEC]); // == LDS offset
        const float* glb = lb    + base;
        const float* gub = ub    + base;
        const float* gal = alpha + base;
        asm volatile("global_load_async_to_lds_b128 %0, %1, off"
                     :: "v"(llb), "v"(glb) : "memory");
        asm volatile("global_load_async_to_lds_b128 %0, %1, off"
                     :: "v"(lub), "v"(gub) : "memory");
        asm volatile("global_load_async_to_lds_b128 %0, %1, off"
                     :: "v"(lal), "v"(gal) : "memory");
        asm volatile("s_wait_asynccnt 0x0" ::: "memory");
    }

    // --- compute 4 elements from LDS ---------------------------------------
    const float4 L = *reinterpret_cast<const float4*>(&s_lb[t * VEC]);
    const float4 U = *reinterpret_cast<const float4*>(&s_ub[t * VEC]);
    const float4 A = *reinterpret_cast<const float4*>(&s_al[t * VEC]);

    float4 r_ub, r_lb;
    relu_relax(L.x, U.x, A.x, r_ub.x, r_lb.x);
    relu_relax(L.y, U.y, A.y, r_ub.y, r_lb.y);
    relu_relax(L.z, U.z, A.z, r_ub.z, r_lb.z);
    relu_relax(L.w, U.w, A.w, r_ub.w, r_lb.w);

    // d_out = [out_ub (n floats) | out_lb (n floats)], vectorized b128 stores
    *reinterpret_cast<float4*>(out + base)     = r_ub;
    *reinterpret_cast<float4*>(out + n + base) = r_lb;
}

extern "C" void kernel_launch(void* const* d_in, const int* in_sizes, int n_in,
                              void* d_out, int out_size, void* d_ws, size_t ws_size,
                              hipStream_t stream) {
    const float* lb    = (const float*)d_in[0];
    const float* ub    = (const float*)d_in[1];
    const float* alpha = (const float*)d_in[2];
    float* out = (float*)d_out;
    const int n = in_sizes[0];

    const int grid = (n + TILE - 1) / TILE;   // 16 blocks for N=4096
    relu_verifier_kernel<<<grid, BLK, 0, stream>>>(lb, ub, alpha, out, n);
}